// GlobalAttention_37125697307293
// MI455X (gfx1250) — compile-verified
//
#include <hip/hip_runtime.h>
#include <stdint.h>

#define B_    8
#define SRC_  4096
#define TGT_  1024
#define SDIM_ 256
#define TDIM_ 256

// LDS row strides (padded +4 dwords -> conflict-free 16-row fragment reads)
#define QSTR  (SDIM_ + 4)   // 260
#define PSTR  (SRC_  + 4)   // 4100
#define LDS_FLOATS (16 * QSTR + 16 * PSTR)

typedef __attribute__((ext_vector_type(2))) float v2f;
typedef __attribute__((ext_vector_type(8))) float v8f;

__device__ __forceinline__ v8f wmma_f32(v2f a, v2f b, v8f c) {
  // (neg_a, A, neg_b, B, c_mod, C, reuse_a, reuse_b)
  return __builtin_amdgcn_wmma_f32_16x16x4_f32(false, a, false, b,
                                               (short)0, c, false, false);
}

__global__ void __launch_bounds__(256)
GlobalAttention_fused_kernel(const float* __restrict__ MB,    // (B,SRC,SDIM)
                             const float* __restrict__ SRCp,  // (B,TGT,TDIM)
                             const uint8_t* __restrict__ MASK,// (B,SRC) bool
                             const float* __restrict__ WIN,   // (SDIM,TDIM)
                             float* __restrict__ c_out,       // (B,TGT,SDIM)
                             float* __restrict__ p_out)       // (B,TGT,SRC)
{
  extern __shared__ float lds[];
  float* q_lds = lds;                 // 16 x QSTR
  float* p_lds = lds + 16 * QSTR;     // 16 x PSTR

  const int wg   = blockIdx.x;        // 0..511
  const int b    = wg >> 6;           // / (TGT/16)
  const int t0   = (wg & 63) * 16;

  const int tid  = threadIdx.x;
  const int wave = tid >> 5;          // 0..7
  const int lane = tid & 31;
  const int lr   = lane & 15;         // M (A) / N (B) index within tile
  const int kh   = (lane >> 4) << 1;  // K sub-offset: 0 or 2
  const int mh   = (lane >> 4) << 3;  // C/D row half: 0 or 8

  // ------- Phase 1: q strip = source_tile @ W_in^T (16 x 256) -------------
  // Two output tiles per wave in ONE k-loop: shared A fragment, 2 acc chains.
  {
    const int d0a = wave * 16;          // tile w
    const int d0b = d0a + 128;          // tile w+8
    const float* arow  = SRCp + ((size_t)(b * TGT_ + t0 + lr)) * TDIM_;
    const float* browA = WIN + (size_t)(d0a + lr) * TDIM_; // B[k,n] = W_in[n,k]
    const float* browB = WIN + (size_t)(d0b + lr) * TDIM_;
    v8f accA = {}, accB = {};
    for (int k = 0; k < TDIM_; k += 4) {
      v2f a  = *(const v2f*)(arow + k + kh);
      v2f b0 = *(const v2f*)(browA + k + kh);
      v2f b1 = *(const v2f*)(browB + k + kh);
      accA = wmma_f32(a, b0, accA);
      accB = wmma_f32(a, b1, accB);
    }
#pragma unroll
    for (int i = 0; i < 8; ++i) {
      q_lds[(i + mh) * QSTR + d0a + lr] = accA[i];
      q_lds[(i + mh) * QSTR + d0b + lr] = accB[i];
    }
  }
  __syncthreads();

  // ------- Phase 2: score strip = q_strip @ MB^T (16 x 4096) --------------
  // 32 s-tiles per wave, processed as 16 pairs sharing the A fragment.
  {
    const float* arow = q_lds + (size_t)lr * QSTR;
    for (int u = 0; u < 16; ++u) {
      const int s0a = (wave + 16 * u) * 16;   // tile w+16u
      const int s0b = s0a + 128;              // tile w+16u+8
      const float* browA = MB + ((size_t)(b * SRC_ + s0a + lr)) * SDIM_;
      const float* browB = MB + ((size_t)(b * SRC_ + s0b + lr)) * SDIM_;
      if (u + 1 < 16)  // prefetch next pair's B rows -> global_prefetch_b8
        __builtin_prefetch(MB + ((size_t)(b * SRC_ + s0a + 256 + lr)) * SDIM_, 0, 0);
      v8f accA = {}, accB = {};
      for (int k = 0; k < SDIM_; k += 4) {
        v2f a  = *(const v2f*)(arow + k + kh);     // ds_load_b64 (shared)
        v2f b0 = *(const v2f*)(browA + k + kh);    // global_load_b64
        v2f b1 = *(const v2f*)(browB + k + kh);
        accA = wmma_f32(a, b0, accA);
        accB = wmma_f32(a, b1, accB);
      }
      const int  sca = s0a + lr;
      const int  scb = s0b + lr;
      const bool oka = MASK[(size_t)b * SRC_ + sca] != 0;
      const bool okb = MASK[(size_t)b * SRC_ + scb] != 0;
#pragma unroll
      for (int i = 0; i < 8; ++i) {
        p_lds[(size_t)(i + mh) * PSTR + sca] = oka ? accA[i] : -__builtin_inff();
        p_lds[(size_t)(i + mh) * PSTR + scb] = okb ? accB[i] : -__builtin_inff();
      }
    }
  }
  __syncthreads();

  // ------- Phase 3: masked softmax over SRC (2 rows per wave) -------------
  for (int r = wave * 2; r < wave * 2 + 2; ++r) {
    float* prow = p_lds + (size_t)r * PSTR;
    float m = -__builtin_inff();
    for (int s = lane; s < SRC_; s += 32) m = fmaxf(m, prow[s]);
#pragma unroll
    for (int off = 16; off > 0; off >>= 1) m = fmaxf(m, __shfl_xor(m, off, 32));
    float sum = 0.0f;
    for (int s = lane; s < SRC_; s += 32) {
      float e = __expf(prow[s] - m);
      prow[s] = e;
      sum += e;
    }
#pragma unroll
    for (int off = 16; off > 0; off >>= 1) sum += __shfl_xor(sum, off, 32);
    const float inv = 1.0f / sum;
    float* orow = p_out + ((size_t)(b * TGT_ + t0 + r)) * SRC_;
    for (int s = lane; s < SRC_; s += 32) {
      float p = prow[s] * inv;
      prow[s] = p;      // keep normalized probs in LDS for phase 4
      orow[s] = p;      // coalesced align_vectors output
    }
  }
  __syncthreads();

  // ------- Phase 4: c tile = P(16x4096) @ MB (K = 4096) -------------------
  // Two d-tiles per wave in ONE k-loop: A fragment (LDS) loaded once.
  {
    const int d0a = wave * 16;
    const int d0b = d0a + 128;
    const float* prow = p_lds + (size_t)lr * PSTR;
    v8f accA = {}, accB = {};
    for (int k = 0; k < SRC_; k += 4) {
      v2f a = *(const v2f*)(prow + k + kh);
      const float* bp = MB + ((size_t)(b * SRC_ + k + kh)) * SDIM_ + lr;
      v2f b0, b1;
      b0.x = bp[d0a];          // B[k+kh,   d0a+lr]
      b0.y = bp[SDIM_ + d0a];  // B[k+kh+1, d0a+lr]
      b1.x = bp[d0b];
      b1.y = bp[SDIM_ + d0b];
      accA = wmma_f32(a, b0, accA);
      accB = wmma_f32(a, b1, accB);
    }
#pragma unroll
    for (int i = 0; i < 8; ++i) {
      c_out[((size_t)(b * TGT_ + t0 + i + mh)) * SDIM_ + d0a + lr] = accA[i];
      c_out[((size_t)(b * TGT_ + t0 + i + mh)) * SDIM_ + d0b + lr] = accB[i];
    }
  }
}

extern "C" void kernel_launch(void* const* d_in, const int* in_sizes, int n_in,
                              void* d_out, int out_size, void* d_ws, size_t ws_size,
                              hipStream_t stream) {
  (void)in_sizes; (void)n_in; (void)out_size; (void)d_ws; (void)ws_size;
  const float*   MB   = (const float*)d_in[0];
  const float*   S    = (const float*)d_in[1];
  const uint8_t* M    = (const uint8_t*)d_in[2];
  const float*   W    = (const float*)d_in[3];
  float* c_out = (float*)d_out;                                  // (B,TGT,SDIM)
  float* p_out = (float*)d_out + (size_t)B_ * TGT_ * SDIM_;      // (B,TGT,SRC)

  const size_t shmem = (size_t)LDS_FLOATS * sizeof(float);       // ~272.5 KB < 320 KB
  hipFuncSetAttribute(reinterpret_cast<const void*>(&GlobalAttention_fused_kernel),
                      hipFuncAttributeMaxDynamicSharedMemorySize, (int)shmem);

  GlobalAttention_fused_kernel<<<dim3(B_ * (TGT_ / 16)), dim3(256), shmem, stream>>>(
      MB, S, M, W, c_out, p_out);
}